// microGIF_52493090291811
// MI455X (gfx1250) — compile-verified
//
#include <hip/hip_runtime.h>
#include <math.h>

// Native vector type so __builtin_nontemporal_load/store work directly.
typedef float f4 __attribute__((ext_vector_type(4)));

#define ROWS_PER_BLOCK 8
#define BLOCK_THREADS  256

// ---------------------------------------------------------------------------
// Kernel 1: per-column precompute + the three O(N) vector outputs.
//   es[j]  = epsilon[j]*s[j]                                (slope vs w)
//   b [j]  = v[j] + (E_L[j] - v[j] + I_ext[j])/tau_m[j]     (intercept)
// ---------------------------------------------------------------------------
__global__ void microgif_precol(const float* __restrict__ v,
                                const float* __restrict__ eps,
                                const float* __restrict__ s,
                                const float* __restrict__ theta_v,
                                const float* __restrict__ syn_kernel,
                                const float* __restrict__ tau_m,
                                const float* __restrict__ tau_s,
                                const float* __restrict__ tau_theta,
                                const float* __restrict__ J_theta,
                                const float* __restrict__ E_L,
                                const float* __restrict__ I_ext,
                                float* __restrict__ es_out,
                                float* __restrict__ b_out,
                                float* __restrict__ eps_new,
                                float* __restrict__ theta_new,
                                float* __restrict__ adapt,
                                int n)
{
    int j = blockIdx.x * blockDim.x + threadIdx.x;
    if (j >= n) return;

    float sv = s[j];
    float vv = v[j];
    float tm = tau_m[j];

    es_out[j] = eps[j] * sv;                       // I_syn/tau_m slope (R_M = 1)
    b_out[j]  = vv + (E_L[j] - vv + I_ext[j]) / tm;

    // epsilon_new = (0.5 + 0.5*tanh(tss-DELTA)) * exp(-(tss-DELTA)/tau_s)/tau_s
    float tss = syn_kernel[j] + 1.0f;
    float x   = tss - 2.0f;                        // DELTA = 2.0
    float ts  = tau_s[j];
    eps_new[j] = (0.5f + 0.5f * tanhf(x)) * expf(-x / ts) / ts;

    // adaptation_kernel = J_theta/tau_theta * exp(-tss/tau_theta)
    float tt = tau_theta[j];
    float jt = J_theta[j];
    adapt[j] = jt / tt * expf(-tss / tt);

    // theta_v_new = theta_v + (0 - theta_v + J_theta*s)/tau_theta
    float th = theta_v[j];
    theta_new[j] = th + (-th + jt * sv) / tt;
}

// ---------------------------------------------------------------------------
// Kernel 2: bandwidth-bound stream over the 8192x8192 tile.
//   Per-column params staged once per block into LDS via the CDNA5 async
//   copy path (global_load_async_to_lds_b128, ASYNCcnt), then the hot loop
//   streams w with NT b128 loads and writes lambda / v_new with NT b128
//   stores. global_prefetch_b8 pulls the next block-row toward L2.
// ---------------------------------------------------------------------------
__global__ void __launch_bounds__(BLOCK_THREADS)
microgif_main(const f4* __restrict__ w4,
              const f4* __restrict__ es4,
              const f4* __restrict__ b4,
              const f4* __restrict__ th4,
              f4* __restrict__ lam4,
              f4* __restrict__ vn4,
              int ncol4, int nrows)
{
    // 12 KB: [0..1023]=es, [1024..2047]=b, [2048..3071]=theta
    __shared__ __attribute__((aligned(16))) float smem[3 * BLOCK_THREADS * 4];

    int tid = threadIdx.x;
    int j4  = blockIdx.x * blockDim.x + tid;
    int j4c = (j4 < ncol4) ? j4 : (ncol4 - 1);     // clamp for the async stage

    // --- Async stage: 16 B of each param array per thread -> LDS ----------
    {
        unsigned lds_e = (unsigned)(uintptr_t)&smem[tid * 4];   // LDS byte offset
        unsigned lds_b = lds_e + 4096u;
        unsigned lds_t = lds_e + 8192u;
        unsigned goff  = (unsigned)j4c * 16u;                   // global byte offset
        asm volatile(
            "global_load_async_to_lds_b128 %0, %3, %4\n\t"
            "global_load_async_to_lds_b128 %1, %3, %5\n\t"
            "global_load_async_to_lds_b128 %2, %3, %6\n\t"
            "s_wait_asynccnt 0x0"
            :
            : "v"(lds_e), "v"(lds_b), "v"(lds_t), "v"(goff),
              "s"(es4), "s"(b4), "s"(th4)
            : "memory");
    }
    __syncthreads();                                // all waves' copies visible

    if (j4 >= ncol4) return;

    // Per-column constants: one ds_load_b128 each, held in registers.
    f4 e = *(const f4*)&smem[tid * 4];
    f4 b = *(const f4*)&smem[BLOCK_THREADS * 4 + tid * 4];
    f4 t = *(const f4*)&smem[2 * BLOCK_THREADS * 4 + tid * 4];

    int  r0         = blockIdx.y * ROWS_PER_BLOCK;
    long base       = (long)r0 * ncol4 + j4;
    long rowStep    = (long)ncol4;
    bool doPrefetch = (blockIdx.y + 1) < gridDim.y;

#pragma unroll
    for (int r = 0; r < ROWS_PER_BLOCK; ++r) {
        int row = r0 + r;
        if (row < nrows) {
            long idx = base + (long)r * rowStep;

            // Pull next block-row's lines toward L2 (global_prefetch_b8,
            // speculative: dropped on translation failure).
            if (doPrefetch)
                __builtin_prefetch(&w4[idx + (long)ROWS_PER_BLOCK * rowStep], 0, 0);

            f4 wv = __builtin_nontemporal_load(&w4[idx]);   // b128, TH=NT

            f4 vn = b + wv * e;                 // v_next
            f4 x  = (vn - t) * 0.5f;            // (v_next - theta)/DELTA
            f4 lam;
            lam.x = 0.1f * __expf(x.x);         // C * exp(...)
            lam.y = 0.1f * __expf(x.y);
            lam.z = 0.1f * __expf(x.z);
            lam.w = 0.1f * __expf(x.w);
            f4 vnew = (1.0f - lam) * vn;        // (1 - lambda) * v_next

            __builtin_nontemporal_store(lam,  &lam4[idx]);  // b128, TH=NT
            __builtin_nontemporal_store(vnew, &vn4[idx]);   // b128, TH=NT
        }
    }
}

// ---------------------------------------------------------------------------
// Launch. Inputs in setup_inputs() order:
//  0:w(N*N) 1:v 2:epsilon 3:s 4:theta_v 5:syn_kernel 6:tau_m 7:tau_s
//  8:tau_theta 9:J_theta 10:E_L 11:I_ext
// Outputs concatenated: lambda(N*N), v_new(N*N), eps_new(N), theta_new(N),
// adapt(N).
// ---------------------------------------------------------------------------
extern "C" void kernel_launch(void* const* d_in, const int* in_sizes, int n_in,
                              void* d_out, int out_size, void* d_ws, size_t ws_size,
                              hipStream_t stream)
{
    const float* w          = (const float*)d_in[0];
    const float* v          = (const float*)d_in[1];
    const float* eps        = (const float*)d_in[2];
    const float* s          = (const float*)d_in[3];
    const float* theta_v    = (const float*)d_in[4];
    const float* syn_kernel = (const float*)d_in[5];
    const float* tau_m      = (const float*)d_in[6];
    const float* tau_s      = (const float*)d_in[7];
    const float* tau_theta  = (const float*)d_in[8];
    const float* J_theta    = (const float*)d_in[9];
    const float* E_L        = (const float*)d_in[10];
    const float* I_ext      = (const float*)d_in[11];

    const int  n  = in_sizes[1];          // 8192
    const long nn = (long)n * (long)n;

    float* out       = (float*)d_out;
    float* lam       = out;
    float* vnew      = out + nn;
    float* eps_new   = out + 2 * nn;
    float* theta_new = eps_new + n;
    float* adapt     = theta_new + n;

    // Scratch: per-column slope/intercept (2*N floats = 64 KB).
    float* es_ws = (float*)d_ws;
    float* b_ws  = es_ws + n;

    microgif_precol<<<(n + 255) / 256, 256, 0, stream>>>(
        v, eps, s, theta_v, syn_kernel, tau_m, tau_s, tau_theta,
        J_theta, E_L, I_ext, es_ws, b_ws, eps_new, theta_new, adapt, n);

    const int ncol4 = n / 4;
    dim3 block(BLOCK_THREADS);
    dim3 grid((ncol4 + BLOCK_THREADS - 1) / BLOCK_THREADS,
              (n + ROWS_PER_BLOCK - 1) / ROWS_PER_BLOCK);
    microgif_main<<<grid, block, 0, stream>>>(
        (const f4*)w, (const f4*)es_ws, (const f4*)b_ws, (const f4*)theta_v,
        (f4*)lam, (f4*)vnew, ncol4, n);
}